// EncoderGAT_70136815943926
// MI455X (gfx1250) — compile-verified
//
#include <hip/hip_runtime.h>
#include <math.h>

#define NN 100000              // nodes
#define NE 1600000             // edges (without self loops)
#define ET (NE + NN)           // total messages incl. self loops

typedef float v2f __attribute__((ext_vector_type(2)));
typedef float v8f __attribute__((ext_vector_type(8)));
typedef int   v4i __attribute__((vector_size(16)));

#define GLOBAL_AS __attribute__((address_space(1)))
#define LDS_AS    __attribute__((address_space(3)))

#if defined(__HIP_DEVICE_COMPILE__) && defined(__has_builtin)
#  if __has_builtin(__builtin_amdgcn_global_load_async_to_lds_b128) && \
      __has_builtin(__builtin_amdgcn_s_wait_asynccnt)
#    define USE_ASYNC_LDS 1
#  endif
#endif

// ---------------------------------------------------------------------------
// GEMM  H[M,NC] = X[M,K] @ W[K,NC]   (M = NN, multiple of 16)
// One workgroup = 16 rows; one wave per 16-column tile; f32 WMMA 16x16x4.
// X block staged into LDS via async global->LDS DMA (ASYNCcnt) when available.
// ---------------------------------------------------------------------------
template<int K, int NC>
__global__ __launch_bounds__((NC/16)*32)
void gemm_wmma_f32(const float* __restrict__ X, const float* __restrict__ W,
                   float* __restrict__ H) {
  constexpr int LDSK = K + 4;                 // pad to avoid bank conflicts
  __shared__ float xs[16 * LDSK];
  const int row0 = blockIdx.x * 16;
  const int tid  = threadIdx.x;
  constexpr int NTHR = (NC/16)*32;

  const float* xblk = X + (size_t)row0 * K;   // 16 contiguous rows
#ifdef USE_ASYNC_LDS
  // 16*K floats as 16B packets; each packet stays within one (padded) LDS row.
  constexpr int NV = 16 * K / 4;              // float4 packets
  for (int i = tid; i < NV; i += NTHR) {
    const int r  = i / (K/4);
    const int cv = i % (K/4);
    __builtin_amdgcn_global_load_async_to_lds_b128(
        (GLOBAL_AS v4i*)(xblk + (size_t)i * 4),
        (LDS_AS v4i*)(xs + r * LDSK + cv * 4), 0, 0);
  }
  __builtin_amdgcn_s_wait_asynccnt(0);
  __syncthreads();
#else
  for (int i = tid; i < 16*K; i += NTHR) {
    int r = i / K, c = i % K;
    xs[r*LDSK + c] = xblk[i];
  }
  __syncthreads();
#endif

  const int wave = tid >> 5;
  const int lane = tid & 31;
  const int n0   = wave * 16;
  const int j    = lane & 15;
  const bool hi  = lane >= 16;

  v8f acc = {};
  #pragma unroll 4
  for (int k = 0; k < K; k += 4) {
    const int ka = k + (hi ? 2 : 0);
    v2f a, b;
    a.x = xs[j*LDSK + ka];
    a.y = xs[j*LDSK + ka + 1];
    b.x = W[(size_t)ka * NC + n0 + j];
    b.y = W[(size_t)(ka + 1) * NC + n0 + j];
    acc = __builtin_amdgcn_wmma_f32_16x16x4_f32(false, a, false, b,
                                                (short)0, acc, false, false);
  }

  const int rbase = hi ? 8 : 0;
  #pragma unroll
  for (int r = 0; r < 8; ++r)
    H[(size_t)(row0 + rbase + r) * NC + n0 + j] = acc[r];
}

// ---------------------------------------------------------------------------
// Per-node attention coefficients: a_src[n,h] = <h[n,h,:], att_src[h,:]>
// ---------------------------------------------------------------------------
template<int H>
__global__ void attn_coeff(const float* __restrict__ h,
                           const float* __restrict__ att_s,
                           const float* __restrict__ att_d,
                           float* __restrict__ a_s, float* __restrict__ a_d) {
  int t = blockIdx.x * blockDim.x + threadIdx.x;
  if (t >= NN * H) return;
  const int n = t / H, hd = t % H;
  const float* hp = h + (size_t)n * (H*8) + hd*8;
  const float* sp = att_s + hd*8;
  const float* dp = att_d + hd*8;
  float ss = 0.f, sd = 0.f;
  #pragma unroll
  for (int c = 0; c < 8; ++c) { float v = hp[c]; ss += v*sp[c]; sd += v*dp[c]; }
  a_s[t] = ss; a_d[t] = sd;
}

// ---------------------------------------------------------------------------
// Edge helpers
// ---------------------------------------------------------------------------
__device__ __forceinline__ void edge_nodes(const int* __restrict__ ei, int e,
                                           int& s, int& d) {
  if (e < NE) { s = ei[e]; d = ei[NE + e]; }
  else        { s = d = e - NE; }             // self loop
}
__device__ __forceinline__ float lrelu(float x) { return x > 0.f ? x : 0.2f*x; }
// order-preserving map float -> uint so atomicMax(uint) == float max
__device__ __forceinline__ unsigned fenc(float f) {
  unsigned u = __float_as_uint(f);
  return (u >> 31) ? ~u : (u | 0x80000000u);
}
__device__ __forceinline__ float fdec(unsigned u) {
  return (u >> 31) ? __uint_as_float(u ^ 0x80000000u) : __uint_as_float(~u);
}

template<int H>
__global__ void edge_max(const int* __restrict__ ei,
                         const float* __restrict__ a_s,
                         const float* __restrict__ a_d,
                         unsigned* __restrict__ menc) {
  int t = blockIdx.x * blockDim.x + threadIdx.x;
  if (t >= ET * H) return;
  const int e = t / H, hd = t % H;
  int s, d; edge_nodes(ei, e, s, d);
  float el = lrelu(a_s[s*H + hd] + a_d[d*H + hd]);
  atomicMax(&menc[d*H + hd], fenc(el));
}

template<int H>
__global__ void edge_expsum(const int* __restrict__ ei,
                            const float* __restrict__ a_s,
                            const float* __restrict__ a_d,
                            const unsigned* __restrict__ menc,
                            float* __restrict__ denom) {
  int t = blockIdx.x * blockDim.x + threadIdx.x;
  if (t >= ET * H) return;
  const int e = t / H, hd = t % H;
  int s, d; edge_nodes(ei, e, s, d);
  float el = lrelu(a_s[s*H + hd] + a_d[d*H + hd]);
  atomicAdd(&denom[d*H + hd], expf(el - fdec(menc[d*H + hd])));
}

// one thread per (edge, 4-channel group): b128 load of h[src], 4 atomic adds
template<int H>
__global__ void edge_agg(const int* __restrict__ ei,
                         const float* __restrict__ a_s,
                         const float* __restrict__ a_d,
                         const unsigned* __restrict__ menc,
                         const float* __restrict__ denom,
                         const float* __restrict__ h,
                         float* __restrict__ agg) {
  constexpr int OW = H * 8;
  constexpr int G  = OW / 4;                  // float4 groups per edge
  int t = blockIdx.x * blockDim.x + threadIdx.x;
  if (t >= ET * G) return;
  const int e = t / G, c = (t % G) * 4;
  const int hd = c >> 3;                      // group lies within one head
  int s, d; edge_nodes(ei, e, s, d);
  float el = lrelu(a_s[s*H + hd] + a_d[d*H + hd]);
  float w  = expf(el - fdec(menc[d*H + hd])) / denom[d*H + hd];
  const float4 hv = *(const float4*)(h + (size_t)s * OW + c);
  float* ap = agg + (size_t)d * OW + c;
  atomicAdd(ap + 0, hv.x * w);
  atomicAdd(ap + 1, hv.y * w);
  atomicAdd(ap + 2, hv.z * w);
  atomicAdd(ap + 3, hv.w * w);
}

// in-place, vectorized: agg = elu(agg + bias)
__global__ void finalize_elu4(float* __restrict__ agg,
                              const float* __restrict__ bias,
                              int total4, int ow) {
  int t = blockIdx.x * blockDim.x + threadIdx.x;
  if (t >= total4) return;
  float4 v = ((float4*)agg)[t];
  const float4 b = *(const float4*)(bias + ((t * 4) % ow));
  v.x += b.x; v.y += b.y; v.z += b.z; v.w += b.w;
  v.x = v.x > 0.f ? v.x : expf(v.x) - 1.f;
  v.y = v.y > 0.f ? v.y : expf(v.y) - 1.f;
  v.z = v.z > 0.f ? v.z : expf(v.z) - 1.f;
  v.w = v.w > 0.f ? v.w : expf(v.w) - 1.f;
  ((float4*)agg)[t] = v;
}

// one wave per node: v = agg + bias; out = v / max(||v||_2, eps)
__global__ void finalize_norm(const float* __restrict__ agg,
                              const float* __restrict__ bias,
                              float* __restrict__ out) {
  int t = blockIdx.x * blockDim.x + threadIdx.x;   // NN*32 threads exactly
  float v = agg[t] + bias[t & 31];
  float s = v * v;
  #pragma unroll
  for (int o = 16; o > 0; o >>= 1) s += __shfl_xor(s, o, 32);
  out[t] = v / fmaxf(sqrtf(s), 0.0005f);
}

// ---------------------------------------------------------------------------
extern "C" void kernel_launch(void* const* d_in, const int* in_sizes, int n_in,
                              void* d_out, int out_size, void* d_ws, size_t ws_size,
                              hipStream_t stream) {
  (void)in_sizes; (void)n_in; (void)out_size; (void)ws_size;
  const float* x    = (const float*)d_in[0];
  const int*   ei   = (const int*)  d_in[1];
  const float* W1   = (const float*)d_in[2];
  const float* as1  = (const float*)d_in[3];
  const float* ad1  = (const float*)d_in[4];
  const float* b1   = (const float*)d_in[5];
  const float* W2   = (const float*)d_in[6];
  const float* as2  = (const float*)d_in[7];
  const float* ad2  = (const float*)d_in[8];
  const float* b2   = (const float*)d_in[9];
  float* out = (float*)d_out;

  // --- workspace layout (floats). Zeroed block first (one memset). ---
  float* ws = (float*)d_ws;
  size_t o = 0;
  unsigned* m1 = (unsigned*)(ws + o); o += (size_t)NN * 8;   // 0.8M
  float*    dn1 = ws + o;             o += (size_t)NN * 8;   // 0.8M
  float*    agg1 = ws + o;            o += (size_t)NN * 64;  // 6.4M
  unsigned* m2 = (unsigned*)(ws + o); o += (size_t)NN * 4;   // 0.4M
  float*    dn2 = ws + o;             o += (size_t)NN * 4;   // 0.4M
  float*    agg2 = ws + o;            o += (size_t)NN * 32;  // 3.2M
  const size_t zbytes = o * sizeof(float);                   // 48 MB
  float* h1  = ws + o; o += (size_t)NN * 64;
  float* a1s = ws + o; o += (size_t)NN * 8;
  float* a1d = ws + o; o += (size_t)NN * 8;
  float* h2  = ws + o; o += (size_t)NN * 32;
  float* a2s = ws + o; o += (size_t)NN * 4;
  float* a2d = ws + o; o += (size_t)NN * 4;

  (void)hipMemsetAsync(d_ws, 0, zbytes, stream);   // m floor, denom=0, agg=0

  const int B = 256;
  // ---------------- layer 1: 128 -> 8 heads x 8 ----------------
  gemm_wmma_f32<128,64><<<NN/16, 128, 0, stream>>>(x, W1, h1);
  attn_coeff<8><<<(NN*8 + B-1)/B, B, 0, stream>>>(h1, as1, ad1, a1s, a1d);
  edge_max<8><<<(ET*8 + B-1)/B, B, 0, stream>>>(ei, a1s, a1d, m1);
  edge_expsum<8><<<(ET*8 + B-1)/B, B, 0, stream>>>(ei, a1s, a1d, m1, dn1);
  edge_agg<8><<<(ET*16 + B-1)/B, B, 0, stream>>>(ei, a1s, a1d, m1, dn1, h1, agg1);
  finalize_elu4<<<(NN*16 + B-1)/B, B, 0, stream>>>(agg1, b1, NN*16, 64);

  // ---------------- layer 2: 64 -> 4 heads x 8 ----------------
  gemm_wmma_f32<64,32><<<NN/16, 64, 0, stream>>>(agg1, W2, h2);
  attn_coeff<4><<<(NN*4 + B-1)/B, B, 0, stream>>>(h2, as2, ad2, a2s, a2d);
  edge_max<4><<<(ET*4 + B-1)/B, B, 0, stream>>>(ei, a2s, a2d, m2);
  edge_expsum<4><<<(ET*4 + B-1)/B, B, 0, stream>>>(ei, a2s, a2d, m2, dn2);
  edge_agg<4><<<(ET*8 + B-1)/B, B, 0, stream>>>(ei, a2s, a2d, m2, dn2, h2, agg2);
  finalize_norm<<<(NN*32)/B, B, 0, stream>>>(agg2, b2, out);
}